// APPNPRegression_3504693313563
// MI455X (gfx1250) — compile-verified
//
#include <hip/hip_runtime.h>

// APPNP regression, gfx1250 (MI455X).
// Pipeline: degree -> dinv -> WMMA encoder -> K x (edge scatter + fused rescale) -> readout.

typedef __attribute__((ext_vector_type(2))) float v2f;
typedef __attribute__((ext_vector_type(8))) float v8f;

#define HID 16
#define KITERS 10
#define ONE_MINUS_ALPHA 0.9f
#define ALPHA_F 0.1f

__device__ __forceinline__ float relu_f(float v) { return v > 0.f ? v : 0.f; }

// ---------------- degree / dinv ----------------

__global__ void ak_deg_init(float* __restrict__ deg, int n) {
    int i = blockIdx.x * blockDim.x + threadIdx.x;
    if (i < n) deg[i] = 1.0f;  // self-loop contributes 1
}

__global__ void ak_deg_accum(const int* __restrict__ eidx, float* __restrict__ deg, int E) {
    int e = blockIdx.x * blockDim.x + threadIdx.x;
    if (e < E) atomicAdd(&deg[eidx[(long long)E + e]], 1.0f);  // dst row
}

__global__ void ak_dinv(const float* __restrict__ deg, float* __restrict__ dinv, int n) {
    int i = blockIdx.x * blockDim.x + threadIdx.x;
    if (i < n) dinv[i] = rsqrtf(deg[i]);  // deg >= 1 always
}

// ---------------- encoder: h0 = relu(relu(x W1 + b1) W2 + b2), via WMMA ----------------
// One wave = one 16-node tile. A = h1 tile (16x16, M=node, K=hid), B = W2 (KxN),
// computed as 4 chained V_WMMA_F32_16X16X4_F32. Epilogue fuses bias, relu,
// h0 store, g = dinv*h0 store, agg = 0 init.

__global__ void ak_encoder_wmma(const float* __restrict__ x, const float* __restrict__ W1,
                                const float* __restrict__ b1, const float* __restrict__ W2,
                                const float* __restrict__ b2, const float* __restrict__ dinv,
                                float* __restrict__ h0, float* __restrict__ g,
                                float* __restrict__ agg, int n_tiles) {
    int wave = (blockIdx.x * blockDim.x + threadIdx.x) >> 5;
    int lane = threadIdx.x & 31;
    if (wave >= n_tiles) return;  // wave-uniform: EXEC stays all-1s for WMMA

    int row = lane & 15;   // node-in-tile for A; column index for B/C/D
    int hi  = lane >> 4;   // half-wave select
    int node = wave * 16 + row;
    float xv = x[node];

    v8f acc = {};
#pragma unroll
    for (int s = 0; s < 4; ++s) {
        int k0 = 4 * s + 2 * hi;  // lane<16: K={4s,4s+1}; lane>=16: K={4s+2,4s+3}
        v2f a, b;
        a.x = relu_f(xv * W1[k0]     + b1[k0]);      // h1[node][k0]
        a.y = relu_f(xv * W1[k0 + 1] + b1[k0 + 1]);  // h1[node][k0+1]
        b.x = W2[k0 * HID + row];                    // W2[k0][col]
        b.y = W2[(k0 + 1) * HID + row];              // W2[k0+1][col]
        acc = __builtin_amdgcn_wmma_f32_16x16x4_f32(
            false, a, false, b, (short)0, acc, false, false);
    }

    float bb = b2[row];
#pragma unroll
    for (int i = 0; i < 8; ++i) {
        int r = i + 8 * hi;          // C/D layout: VGPR i holds row i (lanes 0-15) / i+8 (16-31)
        int nr = wave * 16 + r;
        float v = relu_f(acc[i] + bb);
        int idx = nr * HID + row;
        h0[idx]  = v;
        g[idx]   = dinv[nr] * v;
        agg[idx] = 0.f;
    }
}

// Scalar tail for N % 16 nodes (N=100000 -> none, kept for generality).
__global__ void ak_encoder_tail(const float* __restrict__ x, const float* __restrict__ W1,
                                const float* __restrict__ b1, const float* __restrict__ W2,
                                const float* __restrict__ b2, const float* __restrict__ dinv,
                                float* __restrict__ h0, float* __restrict__ g,
                                float* __restrict__ agg, int start, int N) {
    int v = start + blockIdx.x * blockDim.x + threadIdx.x;
    if (v >= N) return;
    float h1[HID];
    float xv = x[v];
#pragma unroll
    for (int k = 0; k < HID; ++k) h1[k] = relu_f(xv * W1[k] + b1[k]);
    float dv = dinv[v];
#pragma unroll
    for (int j = 0; j < HID; ++j) {
        float acc = b2[j];
#pragma unroll
        for (int k = 0; k < HID; ++k) acc += h1[k] * W2[k * HID + j];
        float hv = relu_f(acc);
        int idx = v * HID + j;
        h0[idx] = hv; g[idx] = dv * hv; agg[idx] = 0.f;
    }
}

// ---------------- propagation ----------------
// 4 lanes per edge; each lane moves one float4 of g[src] into agg[dst] via fp32 atomics.

__global__ void ak_edge_scatter(const int* __restrict__ eidx, const float* __restrict__ g,
                                float* __restrict__ agg, int E) {
    long long t = (long long)blockIdx.x * blockDim.x + threadIdx.x;
    int e = (int)(t >> 2);
    if (e >= E) return;
    int p = (int)(t & 3);
    int src = eidx[e];
    int dst = eidx[(long long)E + e];
    const float4 val = *(const float4*)(g + (size_t)src * HID + p * 4);
    float* ad = agg + (size_t)dst * HID + p * 4;
    atomicAdd(ad + 0, val.x);
    atomicAdd(ad + 1, val.y);
    atomicAdd(ad + 2, val.z);
    atomicAdd(ad + 3, val.w);
}

// h' = 0.9*dinv*(agg + g) + 0.1*h0 ; then g <- dinv*h', agg <- 0 (ready for next iter).
__global__ void ak_finalize(float4* __restrict__ agg, float4* __restrict__ g,
                            const float4* __restrict__ h0, const float* __restrict__ dinv,
                            int n4) {
    int t = blockIdx.x * blockDim.x + threadIdx.x;
    if (t >= n4) return;
    int v = t >> 2;
    float dv = dinv[v];
    float4 a = agg[t], gg = g[t], h = h0[t];
    float4 hn;
    hn.x = ONE_MINUS_ALPHA * dv * (a.x + gg.x) + ALPHA_F * h.x;
    hn.y = ONE_MINUS_ALPHA * dv * (a.y + gg.y) + ALPHA_F * h.y;
    hn.z = ONE_MINUS_ALPHA * dv * (a.z + gg.z) + ALPHA_F * h.z;
    hn.w = ONE_MINUS_ALPHA * dv * (a.w + gg.w) + ALPHA_F * h.w;
    float4 gn; gn.x = dv * hn.x; gn.y = dv * hn.y; gn.z = dv * hn.z; gn.w = dv * hn.w;
    g[t] = gn;
    float4 z; z.x = 0.f; z.y = 0.f; z.z = 0.f; z.w = 0.f;
    agg[t] = z;
}

// Last iteration fused with readout: out[v] = h'[v] . W3 + b3
__global__ void ak_finalize_out(const float* __restrict__ agg, const float* __restrict__ g,
                                const float* __restrict__ h0, const float* __restrict__ dinv,
                                const float* __restrict__ W3, const float* __restrict__ b3,
                                float* __restrict__ out, int N) {
    int v = blockIdx.x * blockDim.x + threadIdx.x;
    if (v >= N) return;
    float dv = dinv[v];
    float s = b3[0];
#pragma unroll
    for (int j = 0; j < HID; ++j) {
        float h = ONE_MINUS_ALPHA * dv * (agg[v * HID + j] + g[v * HID + j])
                + ALPHA_F * h0[v * HID + j];
        s += h * W3[j];
    }
    out[v] = s;
}

// ---------------- launcher ----------------

extern "C" void kernel_launch(void* const* d_in, const int* in_sizes, int n_in,
                              void* d_out, int out_size, void* d_ws, size_t ws_size,
                              hipStream_t stream) {
    const float* x  = (const float*)d_in[0];
    const int*  eidx = (const int*)d_in[1];   // (2, E) row-major: [0..E)=src, [E..2E)=dst
    const float* W1 = (const float*)d_in[2];
    const float* b1 = (const float*)d_in[3];
    const float* W2 = (const float*)d_in[4];
    const float* b2 = (const float*)d_in[5];
    const float* W3 = (const float*)d_in[6];
    const float* b3 = (const float*)d_in[7];
    float* out = (float*)d_out;

    const int N = in_sizes[0];
    const int E = in_sizes[1] / 2;

    char* ws = (char*)d_ws;
    size_t off = 0;
    auto carve = [&](size_t bytes) -> char* {
        char* p = ws + off;
        off += (bytes + 255) & ~(size_t)255;
        return p;
    };
    float* deg  = (float*)carve((size_t)N * 4);
    float* dinv = (float*)carve((size_t)N * 4);
    float* h0   = (float*)carve((size_t)N * HID * 4);
    float* g    = (float*)carve((size_t)N * HID * 4);
    float* agg  = (float*)carve((size_t)N * HID * 4);

    const int B = 256;
    ak_deg_init<<<(N + B - 1) / B, B, 0, stream>>>(deg, N);
    ak_deg_accum<<<(E + B - 1) / B, B, 0, stream>>>(eidx, deg, E);
    ak_dinv<<<(N + B - 1) / B, B, 0, stream>>>(deg, dinv, N);

    int tiles = N / 16;
    if (tiles > 0) {
        int waves_per_block = B / 32;
        int blocks = (tiles + waves_per_block - 1) / waves_per_block;
        ak_encoder_wmma<<<blocks, B, 0, stream>>>(x, W1, b1, W2, b2, dinv, h0, g, agg, tiles);
    }
    int rem_start = tiles * 16;
    if (rem_start < N) {
        ak_encoder_tail<<<(N - rem_start + B - 1) / B, B, 0, stream>>>(
            x, W1, b1, W2, b2, dinv, h0, g, agg, rem_start, N);
    }

    long long sthreads = 4LL * E;
    int sblocks = (int)((sthreads + B - 1) / B);
    int n4 = N * 4;  // float4 elements in an (N, 16) array
    for (int it = 0; it < KITERS; ++it) {
        ak_edge_scatter<<<sblocks, B, 0, stream>>>(eidx, g, agg, E);
        if (it < KITERS - 1) {
            ak_finalize<<<(n4 + B - 1) / B, B, 0, stream>>>(
                (float4*)agg, (float4*)g, (const float4*)h0, dinv, n4);
        } else {
            ak_finalize_out<<<(N + B - 1) / B, B, 0, stream>>>(
                agg, g, h0, dinv, W3, b3, out, N);
        }
    }
}